// GATConv_23441931501630
// MI455X (gfx1250) — compile-verified
//
#include <hip/hip_runtime.h>

#define IN_FEATS   256
#define NUM_HEADS  4
#define OUT_FEATS  32
#define HF         (NUM_HEADS * OUT_FEATS)   // 128
#define NEG_SLOPE  0.2f

typedef __attribute__((ext_vector_type(2))) float v2f;
typedef __attribute__((ext_vector_type(8))) float v8f;

// ---------------------------------------------------------------------------
// Kernel 1: h[N,128] = feat[N,256] @ fc_w[128,256]^T  via V_WMMA_F32_16X16X4_F32
// One wave -> 16-node x 128-feature strip. 64 K-steps (K=4 each) x 8 N-tiles.
// A: 16x4 f32, lane l holds row (l&15), K-pair offset 2*(l>>4).
// B: 4x16  f32, lane l holds col (l&15), same K-pair offset; since we need
//    B[k][o] = fc_w[o][k], each B fragment is a contiguous float2 from fc_w.
// D: VGPR r of lane l = D[r + 8*(l>>4)][l&15].
// ---------------------------------------------------------------------------
__global__ __launch_bounds__(128) void gat_gemm_wmma(
    const float* __restrict__ feat, const float* __restrict__ fcw,
    float* __restrict__ hout, int N, int ntiles)
{
  const int lane = threadIdx.x & 31;
  const int wave = threadIdx.x >> 5;
  const int tile = blockIdx.x * 4 + wave;
  if (tile >= ntiles) return;            // whole-wave exit: EXEC stays all-1s for WMMA

  const int half = lane >> 4;            // 0 or 1
  const int l16  = lane & 15;
  const int koff = half * 2;

  int arow = tile * 16 + l16;
  if (arow >= N) arow = N - 1;           // clamp: safe redundant read, stores guarded
  const float* aptr = feat + (size_t)arow * IN_FEATS + koff;

  const float* bptr[8];
#pragma unroll
  for (int nt = 0; nt < 8; ++nt)
    bptr[nt] = fcw + (size_t)(nt * 16 + l16) * IN_FEATS + koff;

  const v8f vzero = {0.f, 0.f, 0.f, 0.f, 0.f, 0.f, 0.f, 0.f};
  v8f acc[8];
#pragma unroll
  for (int nt = 0; nt < 8; ++nt) acc[nt] = vzero;

#pragma unroll 4
  for (int k = 0; k < IN_FEATS; k += 4) {
    v2f a = *(const v2f*)(aptr + k);
#pragma unroll
    for (int nt = 0; nt < 8; ++nt) {
      v2f b = *(const v2f*)(bptr[nt] + k);
      acc[nt] = __builtin_amdgcn_wmma_f32_16x16x4_f32(
          false, a, false, b, (short)0, acc[nt], false, false);
    }
  }

#pragma unroll
  for (int r = 0; r < 8; ++r) {
    const int node = tile * 16 + r + 8 * half;
    if (node < N) {
      float* orow = hout + (size_t)node * HF + l16;
#pragma unroll
      for (int nt = 0; nt < 8; ++nt)
        orow[nt * 16] = acc[nt][r];
    }
  }
}

// ---------------------------------------------------------------------------
// Kernel 2: el[n,h] = h[n,h,:].attn_l[h]; er likewise. 1 thread per (n,head).
// ---------------------------------------------------------------------------
__global__ void gat_scores(const float* __restrict__ h,
                           const float* __restrict__ al,
                           const float* __restrict__ ar,
                           float* __restrict__ el, float* __restrict__ er, int N)
{
  int idx = blockIdx.x * blockDim.x + threadIdx.x;
  if (idx >= N * NUM_HEADS) return;
  const int hd = idx & (NUM_HEADS - 1);
  const float* row = h + (size_t)(idx >> 2) * HF + hd * OUT_FEATS;
  const float* wl = al + hd * OUT_FEATS;
  const float* wr = ar + hd * OUT_FEATS;
  float sl = 0.f, sr = 0.f;
#pragma unroll
  for (int f = 0; f < OUT_FEATS; ++f) {
    float v = row[f];
    sl = fmaf(v, wl[f], sl);
    sr = fmaf(v, wr[f], sr);
  }
  el[idx] = sl;
  er[idx] = sr;
}

// ---------------------------------------------------------------------------
// Kernel 3: init out=0, m=-inf, s=0 (harness poisons buffers with 0xAA).
// ---------------------------------------------------------------------------
__global__ void gat_init(float* __restrict__ out, float* __restrict__ m,
                         float* __restrict__ s, int N)
{
  const float NEGINF = __int_as_float(0xFF800000);
  const int total = N * HF;
  for (int i = blockIdx.x * blockDim.x + threadIdx.x; i < total;
       i += gridDim.x * blockDim.x) {
    out[i] = 0.f;
    if (i < N * NUM_HEADS) { m[i] = NEGINF; s[i] = 0.f; }
  }
}

// float atomic max via sign-magnitude integer ordering (valid with -inf init)
__device__ __forceinline__ void atomicMaxF(float* addr, float val)
{
  if (val >= 0.f)
    atomicMax((int*)addr, __float_as_int(val));
  else
    atomicMin((unsigned int*)addr, __float_as_uint(val));
}

// ---------------------------------------------------------------------------
// Kernel 4: per-(edge,head) segment max:  m[dst] = max(el[src]+er[dst])
// ---------------------------------------------------------------------------
__global__ void gat_edge_max(const float* __restrict__ el,
                             const float* __restrict__ er,
                             const int* __restrict__ src,
                             const int* __restrict__ dst,
                             float* __restrict__ m, int E)
{
  const int total = E * NUM_HEADS;
  for (int i = blockIdx.x * blockDim.x + threadIdx.x; i < total;
       i += gridDim.x * blockDim.x) {
    const int e = i >> 2, hd = i & (NUM_HEADS - 1);
    const int sidx = src[e] * NUM_HEADS + hd;
    const int didx = dst[e] * NUM_HEADS + hd;
    atomicMaxF(&m[didx], el[sidx] + er[didx]);
  }
}

// ---------------------------------------------------------------------------
// Kernel 5: s[dst] += exp(leakyrelu(e - m[dst]))   (native global_atomic_add_f32)
// ---------------------------------------------------------------------------
__global__ void gat_edge_sum(const float* __restrict__ el,
                             const float* __restrict__ er,
                             const int* __restrict__ src,
                             const int* __restrict__ dst,
                             const float* __restrict__ m,
                             float* __restrict__ s, int E)
{
  const int total = E * NUM_HEADS;
  for (int i = blockIdx.x * blockDim.x + threadIdx.x; i < total;
       i += gridDim.x * blockDim.x) {
    const int e = i >> 2, hd = i & (NUM_HEADS - 1);
    const int sidx = src[e] * NUM_HEADS + hd;
    const int didx = dst[e] * NUM_HEADS + hd;
    const float ev = el[sidx] + er[didx] - m[didx];
    const float c  = expf(ev >= 0.f ? ev : NEG_SLOPE * ev);
    atomicAdd(&s[didx], c);
  }
}

// ---------------------------------------------------------------------------
// Kernel 6: out[dst,:] += alpha * h[src,:]. One block per edge, 128 lanes cover
// the 4x32 feature vector: fully coalesced 512B gather + 512B atomic add.
// alpha recomputed from el/er/m/s (all L2-resident) instead of spilling coeff.
// ---------------------------------------------------------------------------
__global__ __launch_bounds__(128) void gat_edge_scatter(
    const float* __restrict__ h, const float* __restrict__ el,
    const float* __restrict__ er, const int* __restrict__ src,
    const int* __restrict__ dst, const float* __restrict__ m,
    const float* __restrict__ s, float* __restrict__ out, int E)
{
  const int t  = threadIdx.x;          // 0..127 -> (head, feature)
  const int hd = t >> 5;
  for (int e = blockIdx.x; e < E; e += gridDim.x) {
    const int sn = src[e], dn = dst[e];
    const int didx = dn * NUM_HEADS + hd;
    const float ev = el[sn * NUM_HEADS + hd] + er[didx] - m[didx];
    const float c  = expf(ev >= 0.f ? ev : NEG_SLOPE * ev);
    const float alpha = c / s[didx];
    atomicAdd(out + (size_t)dn * HF + t, alpha * h[(size_t)sn * HF + t]);
  }
}

// ---------------------------------------------------------------------------
extern "C" void kernel_launch(void* const* d_in, const int* in_sizes, int n_in,
                              void* d_out, int out_size, void* d_ws, size_t ws_size,
                              hipStream_t stream)
{
  const float* feat = (const float*)d_in[0];
  const float* fcw  = (const float*)d_in[1];
  const float* al   = (const float*)d_in[2];
  const float* ar   = (const float*)d_in[3];
  const int*   src  = (const int*)d_in[4];
  const int*   dst  = (const int*)d_in[5];
  float* out = (float*)d_out;

  const int N = in_sizes[0] / IN_FEATS;
  const int E = in_sizes[4];

  // workspace layout: h | el | er | m | s   (~57.6 MB for N=100000)
  float* h  = (float*)d_ws;
  float* el = h  + (size_t)N * HF;
  float* er = el + (size_t)N * NUM_HEADS;
  float* m  = er + (size_t)N * NUM_HEADS;
  float* s  = m  + (size_t)N * NUM_HEADS;

  const int ntiles = (N + 15) / 16;
  gat_gemm_wmma<<<(ntiles + 3) / 4, 128, 0, stream>>>(feat, fcw, h, N, ntiles);
  gat_init<<<2048, 256, 0, stream>>>(out, m, s, N);
  gat_scores<<<(N * NUM_HEADS + 255) / 256, 256, 0, stream>>>(h, al, ar, el, er, N);
  gat_edge_max<<<4096, 256, 0, stream>>>(el, er, src, dst, m, E);
  gat_edge_sum<<<4096, 256, 0, stream>>>(el, er, src, dst, m, s, E);
  gat_edge_scatter<<<E, 128, 0, stream>>>(h, el, er, src, dst, m, s, out, E);
}